// _ProposalLayer_8074538516538
// MI455X (gfx1250) — compile-verified
//
#include <hip/hip_runtime.h>
#include <hip/hip_bf16.h>
#include <math.h>

// ---------------- problem constants (from reference) ----------------
#define B_      8
#define A_      9
#define H_      152
#define W_      152
#define K_      (A_*H_*W_)          // 207936 anchors per image
#define PRE_    6000                // PRE_NMS_TOP_N
#define POST_   300                 // POST_NMS_TOP_N
#define SORTN_  8192                // bitonic sort size (pow2 >= PRE_)
#define NWORD_  188                 // ceil(PRE_/32) words per mask row
#define TI_     375                 // PRE_/16 i-tiles
#define TJ_     188                 // j word-tiles (32 j per tile)
#define FEAT_   16.0f
#define NMS_T_  0.7f

typedef float v2f __attribute__((ext_vector_type(2)));
typedef float v8f __attribute__((ext_vector_type(8)));

// float -> order-preserving unsigned key (larger float -> larger key)
__device__ __forceinline__ unsigned orderedKey(float f) {
    unsigned u = __float_as_uint(f);
    return (u & 0x80000000u) ? ~u : (u | 0x80000000u);
}

// ---------------- kernel 1: decode proposals + score keys ----------------
__global__ void k_proposals(const float* __restrict__ scores_in,
                            const float* __restrict__ deltas_in,
                            const float* __restrict__ im_info,
                            const float* __restrict__ anchors,
                            float4* __restrict__ proposals,
                            unsigned* __restrict__ keys)
{
    int gid = blockIdx.x * blockDim.x + threadIdx.x;
    if (gid >= B_ * K_) return;
    int b   = gid / K_;
    int rem = gid - b * K_;           // rem = (h*W + w)*A + a
    int s   = rem / A_;
    int a   = rem - s * A_;
    int h   = s / W_;
    int w   = s - h * W_;

    // objectness score: scores_in[b, A + a, h, w]
    float sc = scores_in[(((size_t)b * (2 * A_) + A_ + a) * H_ + h) * W_ + w];
    keys[gid] = orderedKey(sc);

    // deltas: bbox_deltas[b, 4a + j, h, w]
    size_t plane = (size_t)H_ * W_;
    size_t dbase = (((size_t)b * (4 * A_) + 4 * a) * H_ + h) * W_ + w;
    float dx = deltas_in[dbase];
    float dy = deltas_in[dbase + plane];
    float dw = deltas_in[dbase + 2 * plane];
    float dh = deltas_in[dbase + 3 * plane];

    float ax1 = anchors[a * 4 + 0] + w * FEAT_;
    float ay1 = anchors[a * 4 + 1] + h * FEAT_;
    float ax2 = anchors[a * 4 + 2] + w * FEAT_;
    float ay2 = anchors[a * 4 + 3] + h * FEAT_;

    float aw = ax2 - ax1 + 1.0f, ah = ay2 - ay1 + 1.0f;
    float cx = ax1 + 0.5f * aw,  cy = ay1 + 0.5f * ah;
    float pcx = dx * aw + cx,    pcy = dy * ah + cy;
    float pw  = expf(dw) * aw,   ph  = expf(dh) * ah;
    float x1 = pcx - 0.5f * pw, y1 = pcy - 0.5f * ph;
    float x2 = pcx + 0.5f * pw, y2 = pcy + 0.5f * ph;

    float wmax = im_info[b * 3 + 1] - 1.0f;
    float hmax = im_info[b * 3 + 0] - 1.0f;
    x1 = fminf(fmaxf(x1, 0.0f), wmax);  y1 = fminf(fmaxf(y1, 0.0f), hmax);
    x2 = fminf(fmaxf(x2, 0.0f), wmax);  y2 = fminf(fmaxf(y2, 0.0f), hmax);
    proposals[gid] = make_float4(x1, y1, x2, y2);
}

// ---------------- kernel 2: per-image exact radix-select of top-PRE_ -------
__global__ void __launch_bounds__(1024)
k_radix_select(const unsigned* __restrict__ keys,
               unsigned* __restrict__ thrKey, unsigned* __restrict__ need,
               unsigned* __restrict__ cntG,   unsigned* __restrict__ cntT)
{
    __shared__ unsigned hist[256];
    __shared__ unsigned sh_pref, sh_rem;
    int b = blockIdx.x, tid = threadIdx.x;
    if (tid == 0) { sh_pref = 0u; sh_rem = PRE_; }
    __syncthreads();
    const unsigned* kk = keys + (size_t)b * K_;
    for (int p = 0; p < 4; ++p) {
        int shift = 24 - 8 * p;
        if (tid < 256) hist[tid] = 0u;
        __syncthreads();
        unsigned pref = sh_pref;
        unsigned hiMask = (p == 0) ? 0u : (0xFFFFFFFFu << (shift + 8));
        for (int t = tid; t < K_; t += 1024) {
            unsigned key = kk[t];
            if ((key & hiMask) == (pref & hiMask))
                atomicAdd(&hist[(key >> shift) & 0xFFu], 1u);
        }
        __syncthreads();
        if (tid == 0) {
            unsigned rem = sh_rem, cum = 0;
            for (int v = 255; v >= 0; --v) {
                unsigned c = hist[v];
                if (cum + c >= rem) {
                    sh_pref = pref | ((unsigned)v << shift);
                    sh_rem  = rem - cum;     // still to take among ties of byte v
                    break;
                }
                cum += c;
            }
        }
        __syncthreads();
    }
    if (tid == 0) {
        thrKey[b] = sh_pref;   // exact threshold key T
        need[b]   = sh_rem;    // # elements == T to take
        cntG[b]   = 0u;        // reset compaction counters every launch
        cntT[b]   = 0u;
    }
}

// ---------------- kernel 3: compact (key, ~idx) pairs ----------------
__global__ void k_compact(const unsigned* __restrict__ keys,
                          const unsigned* __restrict__ thrKey,
                          const unsigned* __restrict__ need,
                          unsigned* __restrict__ cntG, unsigned* __restrict__ cntT,
                          unsigned long long* __restrict__ pairs)
{
    int gid = blockIdx.x * blockDim.x + threadIdx.x;
    if (gid >= B_ * K_) return;
    int b = gid / K_;
    unsigned t   = (unsigned)(gid - b * K_);
    unsigned key = keys[gid];
    unsigned T   = thrKey[b];
    // ~idx in low word => for equal keys, smaller index sorts first (descending)
    unsigned long long comp = ((unsigned long long)key << 32) | (unsigned)(~t);
    if (key > T) {
        unsigned pos = atomicAdd(&cntG[b], 1u);             // ends at PRE_-need
        pairs[(size_t)b * SORTN_ + pos] = comp;
    } else if (key == T) {
        unsigned nd = need[b];
        unsigned tt = atomicAdd(&cntT[b], 1u);
        if (tt < nd) pairs[(size_t)b * SORTN_ + (PRE_ - nd) + tt] = comp;
    }
}

// ---------------- kernel 4: LDS bitonic sort (desc) + gather props_top -----
__global__ void __launch_bounds__(1024)
k_sort_gather(const unsigned long long* __restrict__ pairs,
              const float4* __restrict__ proposals,
              float4* __restrict__ propsTop)
{
    __shared__ unsigned long long arr[SORTN_];   // 64 KB of the 320 KB WGP LDS
    int b = blockIdx.x, tid = threadIdx.x;
    for (int t = tid; t < SORTN_; t += 1024)
        arr[t] = (t < PRE_) ? pairs[(size_t)b * SORTN_ + t] : 0ull;  // 0 = -inf sentinel
    __syncthreads();
    for (int k = 2; k <= SORTN_; k <<= 1) {
        for (int j = k >> 1; j > 0; j >>= 1) {
            for (int t = tid; t < SORTN_ / 2; t += 1024) {
                int i = ((t & ~(j - 1)) << 1) | (t & (j - 1));
                int p = i | j;
                bool desc = ((i & k) == 0);
                unsigned long long x = arr[i], y = arr[p];
                if ((x < y) == desc) { arr[i] = y; arr[p] = x; }
            }
            __syncthreads();
        }
    }
    for (int r = tid; r < PRE_; r += 1024) {
        unsigned idx = ~(unsigned)(arr[r] & 0xFFFFFFFFull);
        propsTop[(size_t)b * PRE_ + r] = proposals[(size_t)b * K_ + idx];
    }
}

// ---------------- kernel 5: pairwise IoU suppression bitmask (WMMA) --------
// One wave handles a 16(i) x 32(j) tile. The IoU denominator needs the
// outer sum area_i + area_j: computed as A(16x4) x B(4x16) rank-2 product via
// v_wmma_f32_16x16x4_f32 (A cols = [area_i, 1, 0, 0], B rows = [1; area_j; 0; 0]).
// Intersection is VALU min/max; bits gathered with wave32 ballots matching the
// documented 16x16 f32 D-matrix layout (VGPR r: M=r lanes0-15, M=r+8 lanes16-31).
__global__ void k_iou_mask(const float4* __restrict__ propsTop,
                           unsigned* __restrict__ mask)
{
    int wid  = blockIdx.x * (blockDim.x >> 5) + (threadIdx.x >> 5);
    int lane = threadIdx.x & 31;
    if (wid >= B_ * TI_ * TJ_) return;          // whole-wave exit: EXEC stays all-1s
    int b   = wid / (TI_ * TJ_);
    int rem = wid - b * (TI_ * TJ_);
    int ti  = rem / TJ_;
    int tw  = rem - ti * TJ_;
    int i0  = ti * 16;
    int j0  = tw * 32;
    int half = lane >> 4;
    int ml   = lane & 15;

    const float4* P = propsTop + (size_t)b * PRE_;

    int jA = j0 + ml;                 // subtile 0 column
    int jB = j0 + 16 + ml;            // subtile 1 column
    float4 bjA = P[jA < PRE_ ? jA : PRE_ - 1];  // clamp: no divergence before WMMA
    float4 bjB = P[jB < PRE_ ? jB : PRE_ - 1];
    float areaJA = (bjA.z - bjA.x + 1.0f) * (bjA.w - bjA.y + 1.0f);
    float areaJB = (bjB.z - bjB.x + 1.0f) * (bjB.w - bjB.y + 1.0f);

    float4 biL = P[i0 + ml];
    float areaI = (biL.z - biL.x + 1.0f) * (biL.w - biL.y + 1.0f);

    // A 16x4: VGPR0 = K0(l0-15)/K2(l16-31), VGPR1 = K1/K3
    v2f a;  a.x  = (half == 0) ? areaI : 0.0f;  a.y  = (half == 0) ? 1.0f   : 0.0f;
    // B 4x16: VGPR0 = row0(l0-15)/row2(l16-31), VGPR1 = row1/row3
    v2f bA; bA.x = (half == 0) ? 1.0f  : 0.0f;  bA.y = (half == 0) ? areaJA : 0.0f;
    v2f bB; bB.x = (half == 0) ? 1.0f  : 0.0f;  bB.y = (half == 0) ? areaJB : 0.0f;
    v8f c = {};
    v8f sumA = __builtin_amdgcn_wmma_f32_16x16x4_f32(false, a, false, bA, (short)0, c, false, false);
    v8f sumB = __builtin_amdgcn_wmma_f32_16x16x4_f32(false, a, false, bB, (short)0, c, false, false);

    unsigned* mrow = mask + ((size_t)b * PRE_ + i0) * NWORD_ + tw;
    #pragma unroll
    for (int r = 0; r < 8; ++r) {
        int M = r + 8 * half;                   // lane's row for this D VGPR
        float4 bm = P[i0 + M];
        // subtile 0
        float xx1 = fmaxf(bm.x, bjA.x), yy1 = fmaxf(bm.y, bjA.y);
        float xx2 = fminf(bm.z, bjA.z), yy2 = fminf(bm.w, bjA.w);
        float inter = fmaxf(xx2 - xx1 + 1.0f, 0.0f) * fmaxf(yy2 - yy1 + 1.0f, 0.0f);
        bool pA = (jA < PRE_) && (inter > NMS_T_ * (sumA[r] - inter));
        // subtile 1
        xx1 = fmaxf(bm.x, bjB.x); yy1 = fmaxf(bm.y, bjB.y);
        xx2 = fminf(bm.z, bjB.z); yy2 = fminf(bm.w, bjB.w);
        inter = fmaxf(xx2 - xx1 + 1.0f, 0.0f) * fmaxf(yy2 - yy1 + 1.0f, 0.0f);
        bool pB = (jB < PRE_) && (inter > NMS_T_ * (sumB[r] - inter));

        unsigned b0 = __builtin_amdgcn_ballot_w32(pA);  // lo16=row r, hi16=row r+8 (j0..j0+15)
        unsigned b1 = __builtin_amdgcn_ballot_w32(pB);  // same rows, j0+16..j0+31
        if (lane == 0) {
            mrow[(size_t)r * NWORD_]       = (b0 & 0xFFFFu) | ((b1 & 0xFFFFu) << 16);
            mrow[(size_t)(r + 8) * NWORD_] = (b0 >> 16)     | (b1 & 0xFFFF0000u);
        }
    }
}

// ---------------- kernel 6: single-wave exact greedy NMS scan --------------
__global__ void k_nms_greedy(const float4* __restrict__ propsTop,
                             const unsigned* __restrict__ mask,
                             float* __restrict__ out)
{
    __shared__ unsigned removed[NWORD_];
    int b = blockIdx.x, lane = threadIdx.x;   // blockDim = 32 (one wave32)
    for (int w = lane; w < NWORD_; w += 32) removed[w] = 0u;
    __syncthreads();
    float* o = out + (size_t)b * POST_ * 5;
    const unsigned* mbase = mask + (size_t)b * PRE_ * NWORD_;
    int kept = 0;
    for (int i = 0; i < PRE_ && kept < POST_; ++i) {
        unsigned wv = removed[i >> 5];        // LDS broadcast -> wave-uniform
        if (!(wv & (1u << (i & 31)))) {
            if (lane == 0) {
                float4 p = propsTop[(size_t)b * PRE_ + i];
                o[kept * 5 + 0] = (float)b;
                o[kept * 5 + 1] = p.x; o[kept * 5 + 2] = p.y;
                o[kept * 5 + 3] = p.z; o[kept * 5 + 4] = p.w;
            }
            const unsigned* row = mbase + (size_t)i * NWORD_;
            for (int w = lane; w < NWORD_; w += 32) removed[w] |= row[w];
            ++kept;
            __syncthreads();
        }
    }
    // reference pads exhausted slots with keep=-1 -> boxes zeroed, batch col kept
    for (int r = kept + lane; r < POST_; r += 32) {
        o[r * 5 + 0] = (float)b;
        o[r * 5 + 1] = 0.0f; o[r * 5 + 2] = 0.0f;
        o[r * 5 + 3] = 0.0f; o[r * 5 + 4] = 0.0f;
    }
}

// ---------------- launcher ----------------
extern "C" void kernel_launch(void* const* d_in, const int* in_sizes, int n_in,
                              void* d_out, int out_size, void* d_ws, size_t ws_size,
                              hipStream_t stream)
{
    const float* scores  = (const float*)d_in[0];
    const float* deltas  = (const float*)d_in[1];
    const float* im_info = (const float*)d_in[2];
    const float* anchors = (const float*)d_in[3];
    float* out = (float*)d_out;                // (B, 300, 5) float32

    // workspace carving (all offsets 16B aligned); total ~71 MB
    char* ws = (char*)d_ws;
    size_t off = 0;
    auto take = [&](size_t bytes) { void* p = ws + off; off = (off + bytes + 15) & ~(size_t)15; return p; };
    float4*             proposals = (float4*)            take((size_t)B_ * K_ * 16);
    unsigned*           keys      = (unsigned*)          take((size_t)B_ * K_ * 4);
    unsigned long long* pairs     = (unsigned long long*)take((size_t)B_ * SORTN_ * 8);
    float4*             propsTop  = (float4*)            take((size_t)B_ * PRE_ * 16);
    unsigned*           mask      = (unsigned*)          take((size_t)B_ * PRE_ * NWORD_ * 4);
    unsigned*           thrKey    = (unsigned*)          take(B_ * 4);
    unsigned*           needArr   = (unsigned*)          take(B_ * 4);
    unsigned*           cntG      = (unsigned*)          take(B_ * 4);
    unsigned*           cntT      = (unsigned*)          take(B_ * 4);
    (void)ws_size; (void)in_sizes; (void)n_in; (void)out_size;

    int nAll = B_ * K_;
    k_proposals<<<(nAll + 255) / 256, 256, 0, stream>>>(scores, deltas, im_info, anchors,
                                                        proposals, keys);
    k_radix_select<<<B_, 1024, 0, stream>>>(keys, thrKey, needArr, cntG, cntT);
    k_compact<<<(nAll + 255) / 256, 256, 0, stream>>>(keys, thrKey, needArr, cntG, cntT, pairs);
    k_sort_gather<<<B_, 1024, 0, stream>>>(pairs, proposals, propsTop);
    int nWaves = B_ * TI_ * TJ_;
    k_iou_mask<<<(nWaves + 7) / 8, 256, 0, stream>>>(propsTop, mask);
    k_nms_greedy<<<B_, 32, 0, stream>>>(propsTop, mask, out);
}